// HEALDecoder_46377056863016
// MI455X (gfx1250) — compile-verified
//
#include <hip/hip_runtime.h>
#include <hip/hip_bf16.h>
#include <stdint.h>

#define N_SEND   49152
#define N_REC    65536
#define N_EDGES  262144
#define D_MODEL  256
#define EDGE_OUT 64
#define LIN_IN   320
#define LIN_OUT  256
#define BATCH    2

typedef __attribute__((ext_vector_type(2))) float v2f;
typedef __attribute__((ext_vector_type(8))) float v8f;

// ---------------------------------------------------------------------------
// Kernel 1: edge MLP  edge_feat = relu(edge_attr @ w1 + b1) @ w2 + b2
// Tiny GEMMs (K=4 then K=64); weights staged in LDS, VALU FMA per edge.
// ---------------------------------------------------------------------------
__global__ void __launch_bounds__(256)
edge_mlp_kernel(const float* __restrict__ edge_attr,
                const float* __restrict__ w1, const float* __restrict__ b1,
                const float* __restrict__ w2, const float* __restrict__ b2,
                float* __restrict__ edge_feat) {
    __shared__ float s_w1[4 * 64];
    __shared__ float s_b1[64];
    __shared__ float s_w2[64 * 64];
    __shared__ float s_b2[64];
    const int tid = threadIdx.x;
    if (tid < 256) s_w1[tid] = w1[tid];
    if (tid < 64) { s_b1[tid] = b1[tid]; s_b2[tid] = b2[tid]; }
    for (int i = tid; i < 64 * 64; i += 256) s_w2[i] = w2[i];
    __syncthreads();

    const int e = blockIdx.x * 256 + tid;
    const float4 a = *reinterpret_cast<const float4*>(edge_attr + (size_t)e * 4);

    float h[64];
#pragma unroll
    for (int j = 0; j < 64; ++j) {
        float v = fmaf(a.x, s_w1[j],
                  fmaf(a.y, s_w1[64 + j],
                  fmaf(a.z, s_w1[128 + j],
                  fmaf(a.w, s_w1[192 + j], s_b1[j]))));
        h[j] = fmaxf(v, 0.0f);
    }
    float* o = edge_feat + (size_t)e * 64;
#pragma unroll 4
    for (int j = 0; j < 64; ++j) {
        float acc = s_b2[j];
#pragma unroll
        for (int k = 0; k < 64; ++k) acc = fmaf(h[k], s_w2[k * 64 + j], acc);
        o[j] = acc;
    }
}

// ---------------------------------------------------------------------------
// Kernel 2: zero the scatter accumulator (graph-safe, run every call)
// ---------------------------------------------------------------------------
__global__ void zero_kernel(float4* __restrict__ p, long n4) {
    long i = (long)blockIdx.x * blockDim.x + threadIdx.x;
    if (i < n4) p[i] = make_float4(0.f, 0.f, 0.f, 0.f);
}

// ---------------------------------------------------------------------------
// Kernel 3: scatter-add  vm[b, receivers[e], :] += concat(x[b, senders[e]], ef[e])
// One block per edge, 320 threads = one accumulator row; coalesced atomics.
// ---------------------------------------------------------------------------
__global__ void __launch_bounds__(320)
scatter_kernel(const float* __restrict__ x,
               const int* __restrict__ senders,
               const int* __restrict__ receivers,
               const float* __restrict__ edge_feat,
               float* __restrict__ vm) {
    const int e = blockIdx.x;
    const int f = threadIdx.x;               // 0..319
    const int s = senders[e];
    const int r = receivers[e];
    const float ef = (f >= D_MODEL) ? edge_feat[(size_t)e * 64 + (f - D_MODEL)] : 0.0f;
#pragma unroll
    for (int b = 0; b < BATCH; ++b) {
        float v = (f < D_MODEL) ? x[((size_t)b * N_SEND + s) * D_MODEL + f] : ef;
        atomicAdd(&vm[((size_t)b * N_REC + r) * LIN_IN + f], v);
    }
}

// ---------------------------------------------------------------------------
// Kernel 4: FFN with V_WMMA_F32_16X16X4_F32
//   out = relu(vm @ w1 + b1) @ w2 + b2
// Block = 128 threads (4 waves), 32 receiver rows per block.
// Per wave: 32x64 output slab = 2 row-tiles x 4 col-tiles = 8 v8f accumulators.
// k-outer loop: per k-step 2 A fragments (LDS) + 4 B fragments (global/L2)
// feed 8 wmma -> 0.25 ds-loads and 1 global b32 load per wmma.
//
// A tile staged LDS-direct with GLOBAL_LOAD_ASYNC_TO_LDS_B128 (ASYNCcnt),
// stride 324 floats: rows 16B-aligned for b128, banks (4m+k)%64 conflict-free.
// Dynamic LDS ~73KB (over the 64KB static limit; WGP allows up to 320KB).
// A layout per ISA: lane L -> M = L%16, K = k0 + 2*(L/16) + v.
// B layout mirrored:  lane L -> N = L%16, K = k0 + 2*(L/16) + v.
// C/D layout: lane L, vgpr v -> row M = v + 8*(L/16), col N = L%16.
// ---------------------------------------------------------------------------
#define ROWS 32
#define LDA 324
#define LDH 257
#define FFN_LDS_BYTES ((ROWS * LDA + ROWS * LDH) * sizeof(float))

__global__ void __launch_bounds__(128)
ffn_kernel(const float* __restrict__ vm,
           const float* __restrict__ w1, const float* __restrict__ b1,
           const float* __restrict__ w2, const float* __restrict__ b2,
           float* __restrict__ out) {
    extern __shared__ float smem[];
    float* sA = smem;                 // ROWS x LDA input tile (padded)
    float* sH = smem + ROWS * LDA;    // ROWS x LDH hidden tile (padded)

    const int tid  = threadIdx.x;
    const int wave = tid >> 5;
    const int lane = tid & 31;
    const int hi   = lane >> 4;       // half-wave: 0 or 1
    const int m    = lane & 15;       // row within tile / col within tile
    const int r0   = blockIdx.x * ROWS;
    const int bb   = blockIdx.y;
    const int nb   = wave * 64;       // this wave's 64-column slab base

    // Stage A = vm[bb, r0:r0+32, 0:320] directly into LDS via async B128 copies.
    // 32*320 floats = 2560 float4 chunks, 20 per thread; no VGPR round trip.
    {
        const float* Arow = vm + ((size_t)bb * N_REC + r0) * LIN_IN;
        const uint32_t lds_base = (uint32_t)(uintptr_t)sA;  // low 32b = LDS offset
#pragma unroll
        for (int j = 0; j < (ROWS * LIN_IN / 4) / 128; ++j) {
            const int i  = tid + 128 * j;         // float4 chunk index
            const int r  = i / (LIN_IN / 4);      // 80 float4 per row
            const int c4 = i - r * (LIN_IN / 4);
            const uint32_t lds_addr = lds_base + (uint32_t)(r * LDA + c4 * 4) * 4u;
            const float* gptr = Arow + (size_t)r * LIN_IN + (size_t)c4 * 4;
            asm volatile("global_load_async_to_lds_b128 %0, %1, off"
                         :: "v"(lds_addr), "v"(gptr)
                         : "memory");
        }
        asm volatile("s_wait_asynccnt 0" ::: "memory");
    }
    __syncthreads();

    // ---- Stage 1: h2 = relu(A @ w1 + b1), K = 320 -------------------------
    {
        v8f acc[2][4];
#pragma unroll
        for (int t = 0; t < 4; ++t) {
            const float bias = b1[nb + t * 16 + m];
#pragma unroll
            for (int rt = 0; rt < 2; ++rt)
#pragma unroll
                for (int v = 0; v < 8; ++v) acc[rt][t][v] = bias;
        }

        for (int k0 = 0; k0 < LIN_IN; k0 += 4) {
            const int kA = k0 + 2 * hi;
            v2f a0, a1;
            a0.x = sA[m * LDA + kA];
            a0.y = sA[m * LDA + kA + 1];
            a1.x = sA[(16 + m) * LDA + kA];
            a1.y = sA[(16 + m) * LDA + kA + 1];
#pragma unroll
            for (int t = 0; t < 4; ++t) {
                const int n = nb + t * 16 + m;
                v2f bv;
                bv.x = w1[(size_t)kA * LIN_OUT + n];
                bv.y = w1[(size_t)(kA + 1) * LIN_OUT + n];
                acc[0][t] = __builtin_amdgcn_wmma_f32_16x16x4_f32(
                    false, a0, false, bv, (short)0, acc[0][t], false, false);
                acc[1][t] = __builtin_amdgcn_wmma_f32_16x16x4_f32(
                    false, a1, false, bv, (short)0, acc[1][t], false, false);
            }
        }
#pragma unroll
        for (int t = 0; t < 4; ++t) {
            const int n = nb + t * 16 + m;
#pragma unroll
            for (int rt = 0; rt < 2; ++rt)
#pragma unroll
                for (int v = 0; v < 8; ++v)
                    sH[(rt * 16 + v + 8 * hi) * LDH + n] = fmaxf(acc[rt][t][v], 0.0f);
        }
    }
    __syncthreads();

    // ---- Stage 2: out = h2 @ w2 + b2, K = 256 -----------------------------
    {
        v8f acc[2][4];
#pragma unroll
        for (int t = 0; t < 4; ++t) {
            const float bias = b2[nb + t * 16 + m];
#pragma unroll
            for (int rt = 0; rt < 2; ++rt)
#pragma unroll
                for (int v = 0; v < 8; ++v) acc[rt][t][v] = bias;
        }

        for (int k0 = 0; k0 < LIN_OUT; k0 += 4) {
            const int kA = k0 + 2 * hi;
            v2f a0, a1;
            a0.x = sH[m * LDH + kA];
            a0.y = sH[m * LDH + kA + 1];
            a1.x = sH[(16 + m) * LDH + kA];
            a1.y = sH[(16 + m) * LDH + kA + 1];
#pragma unroll
            for (int t = 0; t < 4; ++t) {
                const int n = nb + t * 16 + m;
                v2f bv;
                bv.x = w2[(size_t)kA * LIN_OUT + n];
                bv.y = w2[(size_t)(kA + 1) * LIN_OUT + n];
                acc[0][t] = __builtin_amdgcn_wmma_f32_16x16x4_f32(
                    false, a0, false, bv, (short)0, acc[0][t], false, false);
                acc[1][t] = __builtin_amdgcn_wmma_f32_16x16x4_f32(
                    false, a1, false, bv, (short)0, acc[1][t], false, false);
            }
        }

        float* Orow = out + ((size_t)bb * N_REC + r0) * LIN_OUT;
#pragma unroll
        for (int t = 0; t < 4; ++t) {
            const int n = nb + t * 16 + m;
#pragma unroll
            for (int rt = 0; rt < 2; ++rt)
#pragma unroll
                for (int v = 0; v < 8; ++v)
                    Orow[(size_t)(rt * 16 + v + 8 * hi) * LIN_OUT + n] = acc[rt][t][v];
        }
    }
}

// ---------------------------------------------------------------------------
extern "C" void kernel_launch(void* const* d_in, const int* in_sizes, int n_in,
                              void* d_out, int out_size, void* d_ws, size_t ws_size,
                              hipStream_t stream) {
    const float* x         = (const float*)d_in[0];
    const int*   edge_idx  = (const int*)  d_in[1];
    const float* edge_attr = (const float*)d_in[2];
    const float* ee_w1     = (const float*)d_in[3];
    const float* ee_b1     = (const float*)d_in[4];
    const float* ee_w2     = (const float*)d_in[5];
    const float* ee_b2     = (const float*)d_in[6];
    const float* ff_w1     = (const float*)d_in[7];
    const float* ff_b1     = (const float*)d_in[8];
    const float* ff_w2     = (const float*)d_in[9];
    const float* ff_b2     = (const float*)d_in[10];
    float* out = (float*)d_out;

    // workspace: [edge_feat: N_EDGES*64 f32][vm: BATCH*N_REC*320 f32]
    float* edge_feat = (float*)d_ws;
    float* vm = (float*)((char*)d_ws + (size_t)N_EDGES * 64 * sizeof(float));

    const int* senders   = edge_idx;
    const int* receivers = edge_idx + N_EDGES;

    edge_mlp_kernel<<<N_EDGES / 256, 256, 0, stream>>>(
        edge_attr, ee_w1, ee_b1, ee_w2, ee_b2, edge_feat);

    const long n4 = (long)BATCH * N_REC * LIN_IN / 4;
    zero_kernel<<<(unsigned)((n4 + 255) / 256), 256, 0, stream>>>((float4*)vm, n4);

    scatter_kernel<<<N_EDGES, 320, 0, stream>>>(x, senders, receivers, edge_feat, vm);

    dim3 g(N_REC / ROWS, BATCH);
    ffn_kernel<<<g, 128, FFN_LDS_BYTES, stream>>>(vm, ff_w1, ff_b1, ff_w2, ff_b2, out);
}